// Model_72559177499427
// MI455X (gfx1250) — compile-verified
//
#include <hip/hip_runtime.h>
#include <cstdint>
#include <cstddef>

// Problem constants (match reference): B=8, L=256, C=64
#define Lc   256
#define Bc   8
#define Cc   64
#define NEGINF (-1e30f)
// Padded LDS row stride in dwords: column reads s[k*SROW + e] with consecutive
// k per lane hit bank (k*257+e)%64 = (k+e)%64 -> conflict-free.
#define SROW 257
#define LDS_BYTES (Lc * SROW * 4)   // 263168 B <= 320KB WGP LDS

typedef float        v4f __attribute__((ext_vector_type(4)));
typedef unsigned int v4u __attribute__((ext_vector_type(4)));
typedef int          v4i __attribute__((ext_vector_type(4)));
typedef int          v8i __attribute__((ext_vector_type(8)));

// ---------------------------------------------------------------------------
// Phase 1: t[b,i,j] = max_c scores[b,i,j,c].
// 16 lanes cooperate per (b,i,j) cell; each lane loads one float4 (b128),
// so a wave reads 512B fully contiguous. Non-temporal: 134MB streamed once,
// keep L2 free for the 2MB t table that phase 2 re-reads.
// ---------------------------------------------------------------------------
__global__ __launch_bounds__(256)
void reduce_max_c(const float* __restrict__ scores, float* __restrict__ t) {
  const unsigned gid  = blockIdx.x * 256u + threadIdx.x;
  const unsigned cell = gid >> 4;          // (b*L + i)*L + j
  const unsigned l16  = gid & 15u;
  const v4f v = __builtin_nontemporal_load((const v4f*)scores + ((size_t)cell * 16 + l16));
  float m = fmaxf(fmaxf(v.x, v.y), fmaxf(v.z, v.w));
  // reduce within the 16-lane group (xor offsets < 16 stay in-group, wave32)
  m = fmaxf(m, __shfl_xor(m, 8, 32));
  m = fmaxf(m, __shfl_xor(m, 4, 32));
  m = fmaxf(m, __shfl_xor(m, 2, 32));
  m = fmaxf(m, __shfl_xor(m, 1, 32));
  if (l16 == 0) t[cell] = m;
}

// ---------------------------------------------------------------------------
// Phase 2: CKY max-plus inside pass, one workgroup (1024 thr = 32 waves) per
// batch. Whole 256x256 DP table lives in LDS (stride-257 padded). The table
// is preloaded with t via one TDM tensor_load_to_lds whose pad fields
// (pad 1 dword every 256 dwords) generate the 257-stride layout in-flight.
// DP: for w=2..255: s[i][i+w] += max_{i<k<i+w}(s[i][k] + s[k][i+w]).
// (w==1 cells already equal t from the preload.)
// ---------------------------------------------------------------------------
__global__ __launch_bounds__(1024)
void cky_inside(const float* __restrict__ t, const int* __restrict__ lens,
                float* __restrict__ out) {
  extern __shared__ float s[];             // [Lc][SROW]
  const int b   = blockIdx.x;
  const int tid = threadIdx.x;
  const float* __restrict__ tb = t + (size_t)b * Lc * Lc;

#if defined(__gfx1250__) && __has_builtin(__builtin_amdgcn_tensor_load_to_lds)
  // TDM issues regardless of EXEC -> only wave 0 may reach the instruction.
  if (tid < 32) {
    const unsigned long long ga = (unsigned long long)(uintptr_t)tb;
    const unsigned laddr = (unsigned)(uintptr_t)(&s[0]);   // low 32b = LDS offset
    // D# group 0: count=1 (valid), lds_addr, 57b global addr, type=2 (image)
    v4u g0;
    g0.x = 1u;
    g0.y = laddr;
    g0.z = (unsigned)(ga & 0xFFFFFFFFull);
    g0.w = (unsigned)(((ga >> 32) & 0x1FFFFFFull) | 0x80000000u);
    // D# group 1:
    // dw0: data_size=2 (4B), pad_enable=1, pad_interval=7 (256 dw), pad_amount=0 (1 dw)
    // dw1..dw3: tensor_dim0=256, tensor_dim1=256, tile_dim0=256 (each in hi16)
    // dw4: tile_dim1=256 (lo16), tile_dim2=0
    // dw5..dw7: tensor_dim0_stride=256, tensor_dim1_stride=0 (2D tile)
    v8i g1;
    g1[0] = 0x01D20000; g1[1] = 0x01000000; g1[2] = 0x01000000; g1[3] = 0x01000000;
    g1[4] = 0x00000100; g1[5] = 256;        g1[6] = 0;          g1[7] = 0;
    v4i z4 = {0, 0, 0, 0};                  // groups 2/3 unused for 2D tile
    v8i z8 = {0, 0, 0, 0, 0, 0, 0, 0};      // extra group (6-arg toolchain form)
    __builtin_amdgcn_tensor_load_to_lds(g0, g1, z4, z4, z8, 0);
    __builtin_amdgcn_s_wait_tensorcnt(0);   // TENSORcnt==0 before anyone reads LDS
  }
#else
  for (int idx = tid; idx < Lc * Lc; idx += 1024)
    s[(idx >> 8) * SROW + (idx & (Lc - 1))] = tb[idx];
#endif
  __syncthreads();

  const int wv = tid >> 5;                  // wave id 0..31
  const int ln = tid & 31;                  // lane id
  for (int w = 2; w < Lc; ++w) {
    const int ncell = Lc - w;
    if (w <= 5) {
      // Few splits: thread-per-cell, serial split loop.
      for (int i = tid; i < ncell; i += 1024) {
        const int e = i + w;
        float m = NEGINF;
        for (int k = i + 1; k < e; ++k)
          m = fmaxf(m, s[i * SROW + k] + s[k * SROW + e]);
        s[i * SROW + e] += m;
      }
    } else {
      // Wave-per-cell: lanes stride split k; row read lane-contiguous,
      // column read bank-conflict-free via SROW=257.
      for (int i = wv; i < ncell; i += 32) {
        const int e = i + w;
        const float* __restrict__ rowi = &s[i * SROW];
        float m = NEGINF;
        for (int k = i + 1 + ln; k < e; k += 32)
          m = fmaxf(m, rowi[k] + s[k * SROW + e]);
        m = fmaxf(m, __shfl_xor(m, 16, 32));
        m = fmaxf(m, __shfl_xor(m, 8, 32));
        m = fmaxf(m, __shfl_xor(m, 4, 32));
        m = fmaxf(m, __shfl_xor(m, 2, 32));
        m = fmaxf(m, __shfl_xor(m, 1, 32));
        if (ln == 0) s[i * SROW + e] += m;  // span-w cell: no intra-step RAW
      }
    }
    __syncthreads();
  }

  if (tid == 0) {
    int lc = lens[b];
    lc = lc < 1 ? 1 : (lc > Lc - 1 ? Lc - 1 : lc);
    out[b] = s[lc];                         // s[0][lc]
  }
}

// ---------------------------------------------------------------------------
extern "C" void kernel_launch(void* const* d_in, const int* in_sizes, int n_in,
                              void* d_out, int out_size, void* d_ws, size_t ws_size,
                              hipStream_t stream) {
  const float* scores = (const float*)d_in[0];   // [B,L,L,C] f32
  const int*   lens   = (const int*)d_in[1];     // [B] i32
  float*       out    = (float*)d_out;           // [B] f32
  float*       t      = (float*)d_ws;            // [B,L,L] f32 scratch (2MB)

  // Phase 1: B*L*L cells * 16 lanes = 8,388,608 threads -> 32768 blocks.
  reduce_max_c<<<(Bc * Lc * Lc * 16) / 256, 256, 0, stream>>>(scores, t);

  // Allow 257KB of dynamic LDS (one workgroup per WGP).
  (void)hipFuncSetAttribute(reinterpret_cast<const void*>(&cky_inside),
                            hipFuncAttributeMaxDynamicSharedMemorySize, LDS_BYTES);
  cky_inside<<<Bc, 1024, LDS_BYTES, stream>>>(t, lens, out);
}